// GCN_68143951118599
// MI455X (gfx1250) — compile-verified
//
#include <hip/hip_runtime.h>
#include <hip/hip_bf16.h>

// ---------------------------------------------------------------------------
// GCN (2-layer) for MI455X / gfx1250, wave32.
//   h1  = relu( Agg(x@W1) + (x@W1)*dinv^2 + b1 )
//   out =       Agg(h1@W2) + (h1@W2)*dinv^2 + b2
// GEMMs use V_WMMA_F32_16X16X4_F32 (exact fp32). Aggregation uses native
// global f32 atomics (memory/L2-bound part of the workload).
// ---------------------------------------------------------------------------

typedef __attribute__((ext_vector_type(2))) float v2f;
typedef __attribute__((ext_vector_type(4))) float v4f;
typedef __attribute__((ext_vector_type(8))) float v8f;

// Only meaningful during the device pass; host pass has no amdgcn builtins.
#if defined(__HIP_DEVICE_COMPILE__) && !__has_builtin(__builtin_amdgcn_wmma_f32_16x16x4_f32)
#error "missing __builtin_amdgcn_wmma_f32_16x16x4_f32 on this toolchain"
#endif

__device__ __forceinline__ void fadd_atomic(float* p, float v) {
#if defined(__AMDGCN__)
    unsafeAtomicAdd(p, v);   // lowers to global_atomic_add_f32 (no CAS loop)
#else
    atomicAdd(p, v);
#endif
}

// ---------------------------------------------------------------------------
// Degree accumulation: deg[dst[e]] += 1
// ---------------------------------------------------------------------------
__global__ void gcn_deg_kernel(const int* __restrict__ dst, float* __restrict__ deg, int E) {
    int e = blockIdx.x * blockDim.x + threadIdx.x;
    if (e < E) fadd_atomic(&deg[dst[e]], 1.0f);
}

// dinv[i] = rsqrt(deg[i] + 1)   (+1 = self loop)
__global__ void gcn_dinv_kernel(const float* __restrict__ deg, float* __restrict__ dinv, int n) {
    int i = blockIdx.x * blockDim.x + threadIdx.x;
    if (i < n) dinv[i] = rsqrtf(deg[i] + 1.0f);
}

// ---------------------------------------------------------------------------
// fp32 WMMA GEMM:  C[M x N] = A[M x K] * B[K x N]   (K, N compile-time)
// One wave per 16x16 output tile; K stepped by 4 with V_WMMA_F32_16X16X4_F32.
// ISA §7.12.2 layouts (wave32):
//   A 16x4 : vgpr j, lane l -> A[M = l%16][K = j + 2*(l/16)]
//   B 4x16 : vgpr j, lane l -> B[K = j + 2*(l/16)][N = l%16]
//   C 16x16: vgpr j, lane l -> C[M = j + 8*(l/16)][N = l%16]
// Compile-time N folds all A/B strides into 24-bit immediate load offsets
// (max 125*128*4 = 64000 B), removing per-step 64-bit address arithmetic.
// M, N are multiples of 16 so EXEC is all-ones for WMMA.
// ---------------------------------------------------------------------------
template <int K, int N>
__global__ void gcn_gemm_wmma_f32(const float* __restrict__ A, const float* __restrict__ B,
                                  float* __restrict__ C, int tileCount) {
    const int lane = threadIdx.x & 31;
    const int wave = threadIdx.x >> 5;
    const int tile = blockIdx.x * (blockDim.x >> 5) + wave;
    if (tile >= tileCount) return;          // wave-uniform: full wave exits together

    constexpr int ntn = N / 16;
    const int tm   = tile / ntn;
    const int tn   = tile - tm * ntn;
    const int half = lane >> 4;             // 0: lanes 0-15, 1: lanes 16-31
    const int lm   = lane & 15;

    const float* arow = A + (size_t)(tm * 16 + lm) * K + 2 * half;   // A[row][2*half + ...]
    const float* bptr = B + (size_t)(2 * half) * N + tn * 16 + lm;   // B[2*half + ...][col]

    v8f c = {};
#if defined(__HIP_DEVICE_COMPILE__)
#pragma unroll
    for (int k0 = 0; k0 < K; k0 += 4) {
        v2f a, b;
        a.x = arow[k0];                     // b64 load, immediate offset
        a.y = arow[k0 + 1];
        b.x = bptr[k0 * N];                 // b32 loads, immediate offsets
        b.y = bptr[(k0 + 1) * N];
        // 8 args: (neg_a, A, neg_b, B, c_mod, C, reuse_a, reuse_b)
        c = __builtin_amdgcn_wmma_f32_16x16x4_f32(false, a, false, b, (short)0, c,
                                                  false, false);
    }
#endif

    float* crow = C + (size_t)(tm * 16 + 8 * half) * N + tn * 16 + lm;
#pragma unroll
    for (int j = 0; j < 8; ++j) crow[j * N] = c[j];
}

// ---------------------------------------------------------------------------
// Edge aggregation: agg[dst] += h[src] * (dinv[src]*dinv[dst])
// One wave per edge; each lane handles F/32 consecutive features
// (vectorized b128/b64 gather + native f32 atomics).
// ---------------------------------------------------------------------------
template <int F>
__global__ void gcn_edge_agg(const int* __restrict__ src, const int* __restrict__ dst,
                             const float* __restrict__ dinv, const float* __restrict__ h,
                             float* __restrict__ agg, int E) {
    const int lane = threadIdx.x & 31;
    const int e    = blockIdx.x * (blockDim.x >> 5) + (threadIdx.x >> 5);
    if (e >= E) return;                     // wave-uniform

    const int   s = src[e];
    const int   d = dst[e];
    const float w = dinv[s] * dinv[d];

    constexpr int V = F / 32;               // floats per lane (4 for F=128, 2 for F=64)
    const float* hp = h   + (size_t)s * F + lane * V;
    float*       ap = agg + (size_t)d * F + lane * V;

    if constexpr (V == 4) {
        v4f hv = *(const v4f*)hp;           // 16B aligned: s*512B + lane*16B
        fadd_atomic(ap + 0, hv.x * w);
        fadd_atomic(ap + 1, hv.y * w);
        fadd_atomic(ap + 2, hv.z * w);
        fadd_atomic(ap + 3, hv.w * w);
    } else {
        v2f hv = *(const v2f*)hp;           // 8B aligned
        fadd_atomic(ap + 0, hv.x * w);
        fadd_atomic(ap + 1, hv.y * w);
    }
}

// ---------------------------------------------------------------------------
// Combine: out[i][f] = (maybe relu)( agg[i][f] + h[i][f]*dinv[i]^2 + bias[f] )
// In-place on `agg`.
// ---------------------------------------------------------------------------
template <int F, bool RELU>
__global__ void gcn_combine(float* __restrict__ agg, const float* __restrict__ h,
                            const float* __restrict__ dinv, const float* __restrict__ bias,
                            int n) {
    int idx = blockIdx.x * blockDim.x + threadIdx.x;
    if (idx >= n * F) return;
    const int   i  = idx / F;
    const int   f  = idx - i * F;
    const float di = dinv[i];
    float v = agg[idx] + h[idx] * di * di + bias[f];
    if (RELU) v = v > 0.0f ? v : 0.0f;
    agg[idx] = v;
}

// ---------------------------------------------------------------------------
// Launch
// ---------------------------------------------------------------------------
extern "C" void kernel_launch(void* const* d_in, const int* in_sizes, int n_in,
                              void* d_out, int out_size, void* d_ws, size_t ws_size,
                              hipStream_t stream) {
    constexpr int IN_DIM = 128, HID_DIM = 128, OUT_DIM = 64;

    const float* x    = (const float*)d_in[0];
    const int*   edge = (const int*)d_in[1];   // int32 (JAX default x64 disabled)
    const float* W1   = (const float*)d_in[2];
    const float* b1   = (const float*)d_in[3];
    const float* W2   = (const float*)d_in[4];
    const float* b2   = (const float*)d_in[5];
    float*       out  = (float*)d_out;

    const int N = in_sizes[0] / IN_DIM;        // 20000 nodes (multiple of 16)
    const int E = in_sizes[1] / 2;             // 640000 edges
    const int* src = edge;
    const int* dst = edge + E;

    // Workspace layout (floats): deg | dinv | h1/h2 | agg1(hrelu)
    float* ws = (float*)d_ws;
    const size_t nPad = ((size_t)N + 511) & ~(size_t)511;
    float* deg  = ws;
    float* dinv = ws + nPad;
    float* h1   = ws + 2 * nPad;                    // also reused for h2
    float* agg1 = h1 + (size_t)N * HID_DIM;         // also holds relu'd hidden

    const int TPB = 256;                            // 8 waves / block

    // ---- degree / normalization -------------------------------------------
    (void)hipMemsetAsync(deg, 0, (size_t)N * sizeof(float), stream);
    gcn_deg_kernel<<<(E + TPB - 1) / TPB, TPB, 0, stream>>>(dst, deg, E);
    gcn_dinv_kernel<<<(N + TPB - 1) / TPB, TPB, 0, stream>>>(deg, dinv, N);

    // ---- layer 1: h1 = x @ W1 ---------------------------------------------
    {
        int tiles = (N / 16) * (HID_DIM / 16);
        int blocks = (tiles + (TPB / 32) - 1) / (TPB / 32);
        gcn_gemm_wmma_f32<IN_DIM, HID_DIM><<<blocks, TPB, 0, stream>>>(x, W1, h1, tiles);
    }
    (void)hipMemsetAsync(agg1, 0, (size_t)N * HID_DIM * sizeof(float), stream);
    gcn_edge_agg<HID_DIM><<<(E + (TPB / 32) - 1) / (TPB / 32), TPB, 0, stream>>>(
        src, dst, dinv, h1, agg1, E);
    gcn_combine<HID_DIM, true><<<((size_t)N * HID_DIM + TPB - 1) / TPB, TPB, 0, stream>>>(
        agg1, h1, dinv, b1, N);

    // ---- layer 2: h2 = hrelu @ W2 (h2 reuses h1 buffer) -------------------
    {
        int tiles = (N / 16) * (OUT_DIM / 16);
        int blocks = (tiles + (TPB / 32) - 1) / (TPB / 32);
        gcn_gemm_wmma_f32<HID_DIM, OUT_DIM><<<blocks, TPB, 0, stream>>>(agg1, W2, h1, tiles);
    }
    (void)hipMemsetAsync(out, 0, (size_t)N * OUT_DIM * sizeof(float), stream);
    gcn_edge_agg<OUT_DIM><<<(E + (TPB / 32) - 1) / (TPB / 32), TPB, 0, stream>>>(
        src, dst, dinv, h1, out, E);
    gcn_combine<OUT_DIM, false><<<((size_t)N * OUT_DIM + TPB - 1) / TPB, TPB, 0, stream>>>(
        out, h1, dinv, b2, N);
}